// SimpleSCINet_83004537962692
// MI455X (gfx1250) — compile-verified
//
#include <hip/hip_runtime.h>
#include <math.h>

typedef __attribute__((ext_vector_type(2))) float v2f;
typedef __attribute__((ext_vector_type(4))) float v4f;
typedef __attribute__((ext_vector_type(8))) float v8f;

// Explicit address spaces: LDS = 3, global = 1. This guarantees ds_* / global_*
// lowering (no flat_load, no per-access generic->LDS null checks).
#define AS3 __attribute__((address_space(3)))
#define AS1 __attribute__((address_space(1)))
typedef AS3 float*       lptr;
typedef const AS3 float* clptr;
typedef AS1 float*       gptr;
typedef const AS1 float* cgptr;

#define B_TOT    8192
#define L_SEQ    256
#define NCH      19
#define KPACK1   96     // conv1 K: 19*5=95 -> 96 (zero padded)
#define KPACK2   64     // conv2 K: 19*3=57 -> 64 (zero padded)
#define NPACK    32     // out channels 19 -> 32 (zero padded)
#define PRS      96     // LDS pair-row stride (floats): +1 pair-row lands on the
                        // opposite 32-bank half -> lanes 0-15 vs 16-31 conflict-free
#define NWAVES   8
#define NTHREADS 256
#define K1N      (KPACK1 * NPACK)   // 3072 floats per branch (conv1)
#define K2N      (KPACK2 * NPACK)   // 2048 floats per branch (conv2)

// LDS layout (floats). Worst case Lp=128. +32 slack per buffer so clamped
// A-fragment reads past the valid region hit zeroed LDS (paired with zero
// weights in the padded K region -> contributes exactly 0, never NaN).
#define SEV_OFF 0
#define SEV_SZ  2578                 // (128+6)*19 + 32
#define SOD_OFF (SEV_OFF + SEV_SZ)
#define SOD_SZ  2578
#define HB_OFF  (SOD_OFF + SOD_SZ)
#define HB_SZ   2502                 // (128+2)*19 + 32
#define DB_OFF  (HB_OFF + HB_SZ)
#define DB_SZ   2578
#define CB_OFF  (DB_OFF + DB_SZ)
#define CB_SZ   2578
#define W1S_OFF 12816                // 16B aligned
#define W1S_SZ  ((KPACK1 / 2) * PRS)    // 48 pair-rows * 96 = 4608
#define W2S_OFF (W1S_OFF + W1S_SZ)      // 17424, 16B aligned
#define W2S_SZ  ((KPACK2 / 2) * PRS)    // 32 pair-rows * 96 = 3072
#define SMEM_TOT (W2S_OFF + W2S_SZ)     // 20496 floats = ~80 KB (WGP has 320 KB)

static __device__ __forceinline__ v8f wmma4(v2f a, v2f b, v8f c) {
  // V_WMMA_F32_16X16X4_F32: fp32 matrix pipe, 16x16 tile, K=4
  return __builtin_amdgcn_wmma_f32_16x16x4_f32(false, a, false, b, (short)0, c,
                                               false, false);
}

static __device__ __forceinline__ float leaky(float x) {
  return x > 0.f ? x : 0.01f * x;
}

// tanh on the transcendental unit: clamp(|x|<=15) keeps exp finite, then
// (e-1)/(e+1). ~6 VALU ops vs ~25 for ocml tanhf; correct limits at +-1.
static __device__ __forceinline__ float fast_tanh(float x) {
  x = fminf(fmaxf(x, -15.f), 15.f);
  const float e = __expf(2.f * x);
  return __fdividef(e - 1.f, e + 1.f);
}

// One lifting branch: replication-padded input (LDS, rows at [3, Lp+3)) ->
// conv1(k=5) as GEMM (M=Lp+2, K=96, N=32) -> leaky -> conv2(k=3) as GEMM
// (M=Lp, K=64, N=32) -> fused epilogue:
//   MODE 0 (phi): outLds = other * exp(tanh(v))      (d = od*exp(phi(ev)))
//   MODE 1 (psi): outLds = other * exp(tanh(v))      (c = ev*exp(psi(od)))
//   MODE 2 (U):   global  = other + tanh(v)          (eu = c + U(d))
//   MODE 3 (P):   global  = other - tanh(v)          (ou = d - P(c))
// Weights live in LDS in K-pair-interleaved form: (k,co) -> pair row k>>1,
// element co*2 + (k&1). A B-fragment {B[kk,n], B[kk+1,n]} (kk even) is then one
// aligned ds_load_b64 (the two N-tiles even fuse into one ds_load_2addr_b64).
template <int MODE>
static __device__ void run_branch(cgptr Wt1, cgptr Wt2, cgptr b1, cgptr b2,
                                  clptr in, lptr hb, lptr w1s, lptr w2s,
                                  int Lp, clptr other, lptr outLds,
                                  gptr xg, int tmul, int tadd) {
  const int tid  = threadIdx.x;
  const int wave = tid >> 5;
  const int lane = tid & 31;
  const int lhi  = lane >> 4;   // half-wave select (K pair / M+8 row)
  const int lln  = lane & 15;   // row (A) / column (B, C/D)

  // ---- stage this branch's packed weights (already pair-interleaved in
  //      global) into LDS with pair-row stride PRS ----
  for (int i = tid; i < K1N / 4; i += NTHREADS) {       // 768 x v4f
    const int p = i >> 4, j = (i & 15) * 4;             // 16 chunks per pair-row
    *(AS3 v4f*)(w1s + p * PRS + j) = ((const AS1 v4f*)Wt1)[i];
  }
  for (int i = tid; i < K2N / 4; i += NTHREADS) {       // 512 x v4f
    const int p = i >> 4, j = (i & 15) * 4;
    *(AS3 v4f*)(w2s + p * PRS + j) = ((const AS1 v4f*)Wt2)[i];
  }
  __syncthreads();

  // ---- GEMM1: h = leaky( window(in) [M1 x 96] * Wt1 [96 x 32] + b1 ) ----
  const int M1   = Lp + 2;
  const int nMt1 = (M1 + 15) >> 4;
  const float b1a = b1[lln];
  const float b1b = b1[lln < 3 ? 16 + lln : 18];        // clamped, no divergence
  for (int mt = wave; mt < nMt1; mt += NWAVES) {
    const int m0 = mt << 4;
    int mrow = m0 + lln; if (mrow > M1 - 1) mrow = M1 - 1;  // clamp
    clptr ap = in + mrow * NCH + 2 * lhi;  // A[m,k] = padded[m*19 + k]
    clptr bp = w1s + lhi * PRS + 2 * lln;  // advances 2 pair-rows per k-step
    v8f acc0 = {}; v8f acc1 = {};
#pragma unroll 4
    for (int k0 = 0; k0 < KPACK1; k0 += 4) {
      v2f a; a.x = ap[k0]; a.y = ap[k0 + 1];
      v2f bv0 = *(const AS3 v2f*)(bp);          // ds_load_b64
      v2f bv1 = *(const AS3 v2f*)(bp + 32);     // ds_load_b64 (co 16..31)
      bp += 2 * PRS;
      acc0 = wmma4(a, bv0, acc0);
      acc1 = wmma4(a, bv1, acc1);
    }
    lptr hrow = hb + (m0 + 8 * lhi) * NCH;
    if (m0 + 16 <= M1) {                  // full tile: no guards, no exec churn
#pragma unroll
      for (int v = 0; v < 8; v++) hrow[v * NCH + lln] = leaky(acc0[v] + b1a);
      if (lln < 3) {
#pragma unroll
        for (int v = 0; v < 8; v++)
          hrow[v * NCH + 16 + lln] = leaky(acc1[v] + b1b);
      }
    } else {                              // last (partial) tile only
#pragma unroll
      for (int v = 0; v < 8; v++)
        if (m0 + 8 * lhi + v < M1) hrow[v * NCH + lln] = leaky(acc0[v] + b1a);
      if (lln < 3) {
#pragma unroll
        for (int v = 0; v < 8; v++)
          if (m0 + 8 * lhi + v < M1)
            hrow[v * NCH + 16 + lln] = leaky(acc1[v] + b1b);
      }
    }
  }
  __syncthreads();

  // ---- GEMM2: o = window(h) [M2 x 64] * Wt2 [64 x 32] + b2, fused epilogue --
  const int M2   = Lp;                 // always a multiple of 16 -> no guards
  const int nMt2 = M2 >> 4;
  const float b2a = b2[lln];
  const float b2b = b2[lln < 3 ? 16 + lln : 18];
  for (int mt = wave; mt < nMt2; mt += NWAVES) {
    const int m0 = mt << 4;
    clptr ap = hb + (m0 + lln) * NCH + 2 * lhi;
    clptr bp = w2s + lhi * PRS + 2 * lln;
    v8f acc0 = {}; v8f acc1 = {};
#pragma unroll 4
    for (int k0 = 0; k0 < KPACK2; k0 += 4) {
      v2f a; a.x = ap[k0]; a.y = ap[k0 + 1];
      v2f bv0 = *(const AS3 v2f*)(bp);
      v2f bv1 = *(const AS3 v2f*)(bp + 32);
      bp += 2 * PRS;
      acc0 = wmma4(a, bv0, acc0);
      acc1 = wmma4(a, bv1, acc1);
    }
    const int mb = m0 + 8 * lhi;
#pragma unroll
    for (int v = 0; v < 8; v++) {
      const int m = mb + v;
      const float t   = fast_tanh(acc0[v] + b2a);
      const float oth = other[(3 + m) * NCH + lln];
      if (MODE <= 1)      outLds[(3 + m) * NCH + lln] = oth * __expf(t);
      else if (MODE == 2) xg[(m * tmul + tadd) * NCH + lln] = oth + t;
      else                xg[(m * tmul + tadd) * NCH + lln] = oth - t;
    }
    if (lln < 3) {                        // single exec toggle for tile 1
      const int co = 16 + lln;
#pragma unroll
      for (int v = 0; v < 8; v++) {
        const int m = mb + v;
        const float t   = fast_tanh(acc1[v] + b2b);
        const float oth = other[(3 + m) * NCH + co];
        if (MODE <= 1)      outLds[(3 + m) * NCH + co] = oth * __expf(t);
        else if (MODE == 2) xg[(m * tmul + tadd) * NCH + co] = oth + t;
        else                xg[(m * tmul + tadd) * NCH + co] = oth - t;
      }
    }
  }
  __syncthreads();
}

// One tree level, in place. Subsequence `sel` (low `depth` bits of t) lives at
// t = i*2^depth + sel; even/odd split is the next bit, so the in-place layout
// after 3 levels equals the reference's recursive interleave order.
__global__ __launch_bounds__(NTHREADS) void scinet_level_kernel(
    float* __restrict__ xbuf_, const float* __restrict__ Wt1p_,
    const float* __restrict__ Wt2p_, const float* __restrict__ b1g_,
    const float* __restrict__ b2g_, int depth) {
  __shared__ __align__(16) float smem[SMEM_TOT];
  const int tid = threadIdx.x;
  const int sel = blockIdx.x & ((1 << depth) - 1);
  const int b   = blockIdx.x >> depth;
  int node = 0;
  for (int i = 0; i < depth; i++) node = 2 * node + 1 + ((sel >> i) & 1);
  const int stride = 1 << depth;
  const int Lp     = (L_SEQ >> depth) >> 1;

  lptr sev = (lptr)(smem + SEV_OFF);
  lptr sod = (lptr)(smem + SOD_OFF);
  lptr hb  = (lptr)(smem + HB_OFF);
  lptr db  = (lptr)(smem + DB_OFF);
  lptr cb  = (lptr)(smem + CB_OFF);
  lptr w1s = (lptr)(smem + W1S_OFF);
  lptr w2s = (lptr)(smem + W2S_OFF);
  lptr sm  = (lptr)smem;

  for (int i = tid; i < SMEM_TOT; i += NTHREADS) sm[i] = 0.f;
  __syncthreads();

  gptr  xg   = (gptr)(xbuf_ + (size_t)b * L_SEQ * NCH);
  cgptr Wt1p = (cgptr)Wt1p_;
  cgptr Wt2p = (cgptr)Wt2p_;
  cgptr b1g  = (cgptr)b1g_;
  cgptr b2g  = (cgptr)b2g_;

  for (int i = tid; i < Lp * NCH; i += NTHREADS) {       // even/odd split
    const int r = i / NCH, c = i - r * NCH;
    sev[(3 + r) * NCH + c] = xg[((2 * r) * stride + sel) * NCH + c];
    sod[(3 + r) * NCH + c] = xg[((2 * r + 1) * stride + sel) * NCH + c];
  }
  __syncthreads();
  for (int i = tid; i < 2 * 3 * NCH; i += NTHREADS) {    // replication pad
    const int c = i % NCH, r = (i / NCH) % 3;
    lptr buf = (i >= 3 * NCH) ? sod : sev;
    buf[r * NCH + c]            = buf[3 * NCH + c];
    buf[(Lp + 3 + r) * NCH + c] = buf[(Lp + 2) * NCH + c];
  }
  __syncthreads();

  const int nb = node * 4;                   // branches: 0=phi 1=psi 2=P 3=U
  // d = od * exp(phi(ev)) ; c = ev * exp(psi(od))
  run_branch<0>(Wt1p + (size_t)(nb + 0) * K1N, Wt2p + (size_t)(nb + 0) * K2N,
                b1g + (nb + 0) * NCH, b2g + (nb + 0) * NCH,
                sev, hb, w1s, w2s, Lp, sod, db, xg, 0, 0);
  run_branch<1>(Wt1p + (size_t)(nb + 1) * K1N, Wt2p + (size_t)(nb + 1) * K2N,
                b1g + (nb + 1) * NCH, b2g + (nb + 1) * NCH,
                sod, hb, w1s, w2s, Lp, sev, cb, xg, 0, 0);
  for (int i = tid; i < 2 * 3 * NCH; i += NTHREADS) {    // pad d, c
    const int c = i % NCH, r = (i / NCH) % 3;
    lptr buf = (i >= 3 * NCH) ? cb : db;
    buf[r * NCH + c]            = buf[3 * NCH + c];
    buf[(Lp + 3 + r) * NCH + c] = buf[(Lp + 2) * NCH + c];
  }
  __syncthreads();
  // eu = c + U(d) -> bit depth = 0 ; ou = d - P(c) -> bit depth = 1
  run_branch<2>(Wt1p + (size_t)(nb + 3) * K1N, Wt2p + (size_t)(nb + 3) * K2N,
                b1g + (nb + 3) * NCH, b2g + (nb + 3) * NCH,
                db, hb, w1s, w2s, Lp, cb, db, xg, 2 * stride, sel);
  run_branch<3>(Wt1p + (size_t)(nb + 2) * K1N, Wt2p + (size_t)(nb + 2) * K2N,
                b1g + (nb + 2) * NCH, b2g + (nb + 2) * NCH,
                cb, hb, w1s, w2s, Lp, db, cb, xg, 2 * stride, sel + stride);
}

// Pack W1/W2 -> [node*4+branch][pair-row k>>1][co*2 + (k&1)], k = tap*19 + ci,
// zero padded in k (95->96 / 57->64) and co (19->32).
__global__ void pack_weights_kernel(const float* __restrict__ W1,
                                    const float* __restrict__ W2,
                                    float* __restrict__ Wt1p,
                                    float* __restrict__ Wt2p) {
  const int idx = blockIdx.x * blockDim.x + threadIdx.x;
  const int N1 = 28 * K1N, N2 = 28 * K2N;
  if (idx < N1) {
    const int nb = idx / K1N, rem = idx - nb * K1N;
    const int p = rem >> 6, q = rem & 63;       // pair-row, element in row
    const int co = q >> 1, k = 2 * p + (q & 1);
    float v = 0.f;
    if (co < NCH && k < 95) {
      const int tap = k / NCH, ci = k - tap * NCH;
      v = W1[((size_t)(nb * NCH + co) * NCH + ci) * 5 + tap];
    }
    Wt1p[idx] = v;
  } else if (idx < N1 + N2) {
    const int j = idx - N1;
    const int nb = j / K2N, rem = j - nb * K2N;
    const int p = rem >> 6, q = rem & 63;
    const int co = q >> 1, k = 2 * p + (q & 1);
    float v = 0.f;
    if (co < NCH && k < 57) {
      const int tap = k / NCH, ci = k - tap * NCH;
      v = W2[((size_t)(nb * NCH + co) * NCH + ci) * 3 + tap];
    }
    Wt2p[j] = v;
  }
}

__global__ void concat_kernel(const float* __restrict__ x0D,
                              const float* __restrict__ xctrl,
                              float* __restrict__ xbuf) {
  const long long idx = (long long)blockIdx.x * blockDim.x + threadIdx.x;
  if (idx >= (long long)B_TOT * L_SEQ * NCH) return;
  const int c = (int)(idx % NCH);
  const long long bl = idx / NCH;
  xbuf[idx] = (c < 5) ? x0D[bl * 5 + c] : xctrl[bl * 14 + (c - 5)];
}

// out[b,o,c] = sum_l proj[o,l] * (tree[b,l,c] + x0D[b,l,c]), c < 5.
// (~1.6% of total FLOPs -> VALU is fine; proj_w stays L2/L0 hot.)
__global__ void proj_kernel(const float* __restrict__ xbuf,
                            const float* __restrict__ x0D,
                            const float* __restrict__ projw,
                            float* __restrict__ out) {
  __shared__ float yres_s[L_SEQ * 5];
  lptr yres = (lptr)yres_s;
  const int b = blockIdx.x, tid = threadIdx.x;
  cgptr xb = (cgptr)xbuf, x0 = (cgptr)x0D, pj = (cgptr)projw;
  for (int i = tid; i < L_SEQ * 5; i += blockDim.x) {
    const int l = i / 5, c = i - l * 5;
    yres[i] = xb[((size_t)b * L_SEQ + l) * NCH + c] +
              x0[((size_t)b * L_SEQ + l) * 5 + c];
  }
  __syncthreads();
  for (int i = tid; i < 64 * 5; i += blockDim.x) {
    const int o = i / 5, c = i - o * 5;
    cgptr pw = pj + o * L_SEQ;
    float acc = 0.f;
    for (int l = 0; l < L_SEQ; l++) acc = fmaf(pw[l], yres[l * 5 + c], acc);
    ((gptr)out)[(size_t)b * 320 + i] = acc;
  }
}

extern "C" void kernel_launch(void* const* d_in, const int* in_sizes, int n_in,
                              void* d_out, int out_size, void* d_ws,
                              size_t ws_size, hipStream_t stream) {
  const float* x0D   = (const float*)d_in[0];
  const float* xctrl = (const float*)d_in[1];
  const float* W1    = (const float*)d_in[2];
  const float* b1    = (const float*)d_in[3];
  const float* W2    = (const float*)d_in[4];
  const float* b2    = (const float*)d_in[5];
  const float* projw = (const float*)d_in[6];
  float* out = (float*)d_out;

  float* ws   = (float*)d_ws;
  float* xbuf = ws;                                   // (B,256,19) state
  const size_t XB = (size_t)B_TOT * L_SEQ * NCH;
  float* Wt1p = xbuf + XB;                            // 28*96*32
  float* Wt2p = Wt1p + 28 * K1N;                      // 28*64*32

  const int packN = 28 * K1N + 28 * K2N;
  pack_weights_kernel<<<(packN + 255) / 256, 256, 0, stream>>>(W1, W2, Wt1p, Wt2p);
  concat_kernel<<<(int)((XB + 255) / 256), 256, 0, stream>>>(x0D, xctrl, xbuf);
  for (int d = 0; d < 3; d++)
    scinet_level_kernel<<<B_TOT << d, NTHREADS, 0, stream>>>(xbuf, Wt1p, Wt2p,
                                                             b1, b2, d);
  proj_kernel<<<B_TOT, 256, 0, stream>>>(xbuf, x0D, projw, out);
}